// SelfAttention_13142599926141
// MI455X (gfx1250) — compile-verified
//
#include <hip/hip_runtime.h>
#include <math.h>

typedef __attribute__((ext_vector_type(2))) float v2f;
typedef __attribute__((ext_vector_type(8))) float v8f;

#define B_   32
#define D_   4096
#define H_   32
#define HKV_ 8
#define G_   4
#define HD_  128
#define L_   2048   // start_pos + 1

// ---------------------------------------------------------------------------
// C[32 x N] = A[32 x K] * W[K x N]   (row-major, f32, WMMA 16x16x4 f32)
// One BLOCK (256 threads = 8 waves) computes one 32x16 column tile with
// split-K: wave w covers K-range [w*K/8, (w+1)*K/8), partials reduced
// deterministically through LDS. N and K are compile-time so the inner loop
// has a static trip count (pointer-bumped, unrolled, load-clause friendly).
// A-fragment (32-bit A 16x4): lanes 0-15 hold M=lane; VGPR0={K0,K2},
// VGPR1={K1,K3} selected by lane half. B mirrors with N across lanes.
// C/D layout: VGPR i -> M = i + 8*(lane>>4), N = lane&15.
// ---------------------------------------------------------------------------
template <int N, int K>
__global__ __launch_bounds__(256)
void gemm32_wmma_splitk(const float* __restrict__ A,
                        const float* __restrict__ W,
                        float* __restrict__ C) {
  __shared__ float s_part[8 * 512];    // 16 KB: [wave][lane*16 + i]

  const int wave  = threadIdx.x >> 5;
  const int lane  = threadIdx.x & 31;
  const int n0    = blockIdx.x * 16;
  const int lhalf = lane >> 4;         // 0/1 -> K offset 0/2
  const int l16   = lane & 15;         // M row / N col within tile

  constexpr int KCHUNK = K / 8;        // per-wave K slice
  const int kbeg = wave * KCHUNK;

  v8f acc0 = {};                       // rows 0..15
  v8f acc1 = {};                       // rows 16..31

  // pointer-bumped streams (no in-loop address math beyond the bump)
  const float* __restrict__ a0p = A + (size_t)l16 * K        + kbeg + 2 * lhalf;
  const float* __restrict__ a1p = A + (size_t)(16 + l16) * K + kbeg + 2 * lhalf;
  const float* __restrict__ bp  = W + (size_t)(kbeg + 2 * lhalf) * N + n0 + l16;

#pragma unroll 8
  for (int it = 0; it < KCHUNK / 4; ++it) {
    v2f a0 = {a0p[0], a0p[1]};
    v2f a1 = {a1p[0], a1p[1]};
    v2f bb = {bp[0], bp[N]};
    acc0 = __builtin_amdgcn_wmma_f32_16x16x4_f32(false, a0, false, bb,
                                                 (short)0, acc0, false, false);
    acc1 = __builtin_amdgcn_wmma_f32_16x16x4_f32(false, a1, false, bb,
                                                 (short)0, acc1, false, false);
    a0p += 4;
    a1p += 4;
    bp  += (size_t)4 * N;
  }

  // stash this wave's partial tile
  float* p = s_part + wave * 512 + lane * 16;
#pragma unroll
  for (int i = 0; i < 8; ++i) { p[i] = acc0[i]; p[8 + i] = acc1[i]; }
  __syncthreads();

  // deterministic cross-wave reduction: each thread owns 2 of 512 elements
  for (int e = threadIdx.x; e < 512; e += 256) {
    float s = 0.f;
#pragma unroll
    for (int w = 0; w < 8; ++w) s += s_part[w * 512 + e];
    const int lane_e = e >> 4;
    const int i      = e & 15;
    const int half   = lane_e >> 4;
    const int col    = n0 + (lane_e & 15);
    const int row    = (i < 8) ? (half * 8 + i) : (16 + half * 8 + (i - 8));
    C[(size_t)row * N + col] = s;
  }
}

// ---------------------------------------------------------------------------
// RoPE applied in-place to Q (32x4096) and K_new (32x1024).
// Flat pair index p: elements 2p, 2p+1; frequency index = p % 64.
// ---------------------------------------------------------------------------
__global__ void rope_kernel(float* __restrict__ q, float* __restrict__ k,
                            const float* __restrict__ cosv,
                            const float* __restrict__ sinv) {
  const int NQ = B_ * H_ * HD_ / 2;     // 65536 pairs
  const int NK = B_ * HKV_ * HD_ / 2;   // 16384 pairs
  int p = blockIdx.x * blockDim.x + threadIdx.x;
  float* base;
  int pp;
  if (p < NQ)            { base = q; pp = p; }
  else if (p < NQ + NK)  { base = k; pp = p - NQ; }
  else return;
  const int f = pp & 63;
  const float c = cosv[f], s = sinv[f];
  const float xr = base[2 * pp], xi = base[2 * pp + 1];
  base[2 * pp]     = xr * c - xi * s;
  base[2 * pp + 1] = xr * s + xi * c;
}

// ---------------------------------------------------------------------------
// Attention: one block per (b, kv_head). 256 threads (8 waves).
// scores (4 x 2048) staged in LDS; two-pass softmax; coalesced V streaming.
// Position L-1 uses the freshly projected K/V (caches are read-only inputs).
// ---------------------------------------------------------------------------
__global__ __launch_bounds__(256)
void attn_kernel(const float* __restrict__ q,      // [B,H,HD] (RoPE'd)
                 const float* __restrict__ knew,   // [B,HKV,HD] (RoPE'd)
                 const float* __restrict__ vnew,   // [B,HKV,HD]
                 const float* __restrict__ cache_k,// [B,HKV,2048,HD]
                 const float* __restrict__ cache_v,
                 float* __restrict__ out) {        // [B,H,HD]
  __shared__ float s_sc[G_][L_];       // 32 KB
  __shared__ float s_q[G_][HD_];       // 2 KB
  __shared__ float s_red[G_][8];

  const int b   = blockIdx.x >> 3;
  const int kvh = blockIdx.x & 7;
  const int tid  = threadIdx.x;
  const int lane = tid & 31;
  const int wave = tid >> 5;

  // load the 4 query heads of this group
  for (int i = tid; i < G_ * HD_; i += 256) {
    const int g = i >> 7, d = i & 127;
    s_q[g][d] = q[(size_t)b * D_ + (size_t)(kvh * G_ + g) * HD_ + d];
  }
  __syncthreads();

  const float scale = 0.08838834764831845f;  // 1/sqrt(128)
  const float* __restrict__ kbase =
      cache_k + ((size_t)b * HKV_ + kvh) * (size_t)L_ * HD_;

  // ---- Phase 1: scores ----
  for (int l = tid; l < L_; l += 256) {
    const float* __restrict__ krow =
        (l == L_ - 1) ? (knew + ((size_t)b * HKV_ + kvh) * HD_)
                      : (kbase + (size_t)l * HD_);
    if (l + 256 < L_ - 1)
      __builtin_prefetch(kbase + (size_t)(l + 256) * HD_, 0, 0);
    float acc[G_] = {0.f, 0.f, 0.f, 0.f};
    for (int d = 0; d < HD_; d += 4) {
      const float4 kv = *(const float4*)(krow + d);
#pragma unroll
      for (int g = 0; g < G_; ++g)
        acc[g] += kv.x * s_q[g][d]     + kv.y * s_q[g][d + 1]
                + kv.z * s_q[g][d + 2] + kv.w * s_q[g][d + 3];
    }
#pragma unroll
    for (int g = 0; g < G_; ++g) s_sc[g][l] = acc[g] * scale;
  }
  __syncthreads();

  // ---- Phase 2: softmax (max then sum-exp), wave32 shuffles + LDS ----
  float lmax[G_] = {-1e30f, -1e30f, -1e30f, -1e30f};
  for (int l = tid; l < L_; l += 256)
#pragma unroll
    for (int g = 0; g < G_; ++g) lmax[g] = fmaxf(lmax[g], s_sc[g][l]);
#pragma unroll
  for (int off = 16; off > 0; off >>= 1)
#pragma unroll
    for (int g = 0; g < G_; ++g)
      lmax[g] = fmaxf(lmax[g], __shfl_xor(lmax[g], off, 32));
  if (lane == 0)
#pragma unroll
    for (int g = 0; g < G_; ++g) s_red[g][wave] = lmax[g];
  __syncthreads();
  float gmax[G_];
#pragma unroll
  for (int g = 0; g < G_; ++g) {
    float m = s_red[g][0];
#pragma unroll
    for (int w = 1; w < 8; ++w) m = fmaxf(m, s_red[g][w]);
    gmax[g] = m;
  }
  __syncthreads();

  float lsum[G_] = {0.f, 0.f, 0.f, 0.f};
  for (int l = tid; l < L_; l += 256)
#pragma unroll
    for (int g = 0; g < G_; ++g) {
      const float e = __expf(s_sc[g][l] - gmax[g]);
      s_sc[g][l] = e;
      lsum[g] += e;
    }
#pragma unroll
  for (int off = 16; off > 0; off >>= 1)
#pragma unroll
    for (int g = 0; g < G_; ++g) lsum[g] += __shfl_xor(lsum[g], off, 32);
  if (lane == 0)
#pragma unroll
    for (int g = 0; g < G_; ++g) s_red[g][wave] = lsum[g];
  __syncthreads();
  float ginv[G_];
#pragma unroll
  for (int g = 0; g < G_; ++g) {
    float s = 0.f;
#pragma unroll
    for (int w = 0; w < 8; ++w) s += s_red[g][w];
    ginv[g] = 1.0f / s;
  }

  // ---- Phase 3: out = probs @ V (coalesced over d) ----
  const float* __restrict__ vbase =
      cache_v + ((size_t)b * HKV_ + kvh) * (size_t)L_ * HD_;
  const int d  = tid & 127;
  const int g0 = (tid >> 7) * 2;       // threads 0-127 -> g 0,1; 128-255 -> g 2,3
  float o0 = 0.f, o1 = 0.f;
  for (int l = 0; l < L_ - 1; ++l) {
    if ((l & 7) == 0 && l + 8 < L_ - 1)
      __builtin_prefetch(vbase + (size_t)(l + 8) * HD_ + d, 0, 0);
    const float vv = vbase[(size_t)l * HD_ + d];
    o0 += s_sc[g0][l] * vv;
    o1 += s_sc[g0 + 1][l] * vv;
  }
  {
    const float vv = vnew[((size_t)b * HKV_ + kvh) * HD_ + d];
    o0 += s_sc[g0][L_ - 1] * vv;
    o1 += s_sc[g0 + 1][L_ - 1] * vv;
  }
  out[(size_t)b * D_ + (size_t)(kvh * G_ + g0) * HD_ + d]     = o0 * ginv[g0];
  out[(size_t)b * D_ + (size_t)(kvh * G_ + g0 + 1) * HD_ + d] = o1 * ginv[g0 + 1];
}

// ---------------------------------------------------------------------------
extern "C" void kernel_launch(void* const* d_in, const int* in_sizes, int n_in,
                              void* d_out, int out_size, void* d_ws, size_t ws_size,
                              hipStream_t stream) {
  const float* x       = (const float*)d_in[0];
  const float* cache_k = (const float*)d_in[1];
  const float* cache_v = (const float*)d_in[2];
  const float* wq      = (const float*)d_in[3];
  const float* wk      = (const float*)d_in[4];
  const float* wv      = (const float*)d_in[5];
  const float* wo      = (const float*)d_in[6];
  const float* fcos    = (const float*)d_in[7];
  const float* fsin    = (const float*)d_in[8];
  float* out = (float*)d_out;

  float* ws      = (float*)d_ws;
  float* ws_q    = ws;                         // 32*4096
  float* ws_k    = ws_q + B_ * D_;             // 32*1024
  float* ws_v    = ws_k + B_ * HKV_ * HD_;     // 32*1024
  float* ws_attn = ws_v + B_ * HKV_ * HD_;     // 32*4096

  // QKV projections: one block per 16-column tile, split-K over 8 waves
  gemm32_wmma_splitk<D_, D_>
      <<<dim3(D_ / 16), dim3(256), 0, stream>>>(x, wq, ws_q);
  gemm32_wmma_splitk<HKV_ * HD_, D_>
      <<<dim3((HKV_ * HD_) / 16), dim3(256), 0, stream>>>(x, wk, ws_k);
  gemm32_wmma_splitk<HKV_ * HD_, D_>
      <<<dim3((HKV_ * HD_) / 16), dim3(256), 0, stream>>>(x, wv, ws_v);
  // RoPE on Q and new K
  {
    const int pairs = (B_ * H_ * HD_ + B_ * HKV_ * HD_) / 2;  // 81920
    rope_kernel<<<dim3((pairs + 255) / 256), dim3(256), 0, stream>>>(
        ws_q, ws_k, fcos, fsin);
  }
  // Attention: one block per (batch, kv head)
  attn_kernel<<<dim3(B_ * HKV_), dim3(256), 0, stream>>>(
      ws_q, ws_k, ws_v, cache_k, cache_v, ws_attn);
  // Output projection
  gemm32_wmma_splitk<D_, D_>
      <<<dim3(D_ / 16), dim3(256), 0, stream>>>(ws_attn, wo, out);
}